// NasAutoGraphBCell_36816459661708
// MI455X (gfx1250) — compile-verified
//
#include <hip/hip_runtime.h>
#include <math.h>

typedef __attribute__((ext_vector_type(2))) float v2f;
typedef __attribute__((ext_vector_type(8))) float v8f;

#define HID 128
#define OUTD 128

// ---------------------------------------------------------------------------
// One wave computes a 16x16 f32 tile of D = A(16x128) * B(128x16) using
// V_WMMA_F32_16X16X4_F32, iterating K in steps of 4.
//   A: row-major, lda=128; aRow = row index for this lane's M (=lane&15)
//   B: row-major, ldb=128; caller passes B already offset by n0 (column tile)
// Fragment layouts per CDNA5 ISA 7.12.2 (32-bit A 16x4, B 4x16, C/D 16x16).
// ---------------------------------------------------------------------------
__device__ __forceinline__ v8f wave_gemm128(const float* A, int aRow,
                                            const float* B, int l, int half) {
    const float* a_ptr = A + (size_t)aRow * HID + 2 * half;
    const float* b_ptr = B + (size_t)(2 * half) * OUTD + l;
    v8f acc = {};
#pragma unroll
    for (int k0 = 0; k0 < HID; k0 += 4) {
        v2f a, b;
        a.x = a_ptr[k0];
        a.y = a_ptr[k0 + 1];
        b.x = b_ptr[(size_t)k0 * OUTD];
        b.y = b_ptr[(size_t)(k0 + 1) * OUTD];
        acc = __builtin_amdgcn_wmma_f32_16x16x4_f32(
            /*neg_a=*/false, a, /*neg_b=*/false, b,
            /*c_mod=*/(short)0, acc, /*reuse_a=*/false, /*reuse_b=*/false);
    }
    return acc;
}

// ---------------------------------------------------------------------------
// Kernel 1: h = x @ W_pre + b_pre, then t = h @ arma_init_w.
// One block = 16 rows of nodes; 8 waves each own one 16-wide column tile.
// x tile is staged once into LDS (float4) and shared by all 8 waves; the
// resulting h tile lives in LDS for the fused second GEMM.
// ---------------------------------------------------------------------------
__global__ void __launch_bounds__(256)
gemm_pre_init(const float* __restrict__ x, const float* __restrict__ Wpre,
              const float* __restrict__ bpre, const float* __restrict__ Winit,
              float* __restrict__ h, float* __restrict__ t, int N) {
    __shared__ float xTile[16 * HID];
    __shared__ float hTile[16 * HID];
    const int tid = threadIdx.x;
    const int m0 = blockIdx.x * 16;

    // Cooperative float4 staging of the 16x128 x slab (512 float4s).
    for (int i = tid; i < 16 * HID / 4; i += 256) {
        const int rr = i >> 5;           // 32 float4 per row
        const int ff4 = i & 31;
        int grow = m0 + rr;
        if (grow >= N) grow = N - 1;
        ((float4*)xTile)[i] =
            ((const float4*)(x + (size_t)grow * HID))[ff4];
    }
    __syncthreads();

    const int wave = tid >> 5;
    const int lane = tid & 31;
    const int l = lane & 15;
    const int half = lane >> 4;
    const int n0 = wave * 16;

    v8f acc = wave_gemm128(xTile, l, Wpre + n0, l, half);
    const float bias = bpre[n0 + l];
#pragma unroll
    for (int r = 0; r < 8; ++r) {
        const int trow = 8 * half + r;          // row within tile
        const int grow = m0 + trow;             // global row
        const float v = acc[r] + bias;
        hTile[trow * HID + n0 + l] = v;
        if (grow < N) h[(size_t)grow * HID + n0 + l] = v;
    }
    __syncthreads();

    v8f acc2 = wave_gemm128(hTile, l, Winit + n0, l, half);
#pragma unroll
    for (int r = 0; r < 8; ++r) {
        const int grow = m0 + 8 * half + r;
        if (grow < N) t[(size_t)grow * HID + n0 + l] = acc2[r];
    }
}

// ---------------------------------------------------------------------------
// Kernel 2: weighted in-degree and edge count per target node.
// ---------------------------------------------------------------------------
__global__ void __launch_bounds__(256)
deg_cnt_kernel(const int* __restrict__ ei, const float* __restrict__ ew,
               float* __restrict__ deg, float* __restrict__ cnt, int E) {
    const int e = blockIdx.x * blockDim.x + threadIdx.x;
    if (e >= E) return;
    const int c = ei[E + e];                    // col = edge_index[1]
    atomicAdd(&deg[c], ew[e]);
    atomicAdd(&cnt[c], 1.0f);
}

// ---------------------------------------------------------------------------
// Kernel 3: dis = deg > 0 ? rsqrt(max(deg, 1e-12)) : 0
// ---------------------------------------------------------------------------
__global__ void __launch_bounds__(256)
dis_kernel(const float* __restrict__ deg, float* __restrict__ dis, int N) {
    const int i = blockIdx.x * blockDim.x + threadIdx.x;
    if (i >= N) return;
    const float d = deg[i];
    dis[i] = (d > 0.0f) ? rsqrtf(fmaxf(d, 1e-12f)) : 0.0f;
}

// ---------------------------------------------------------------------------
// Kernel 4: edge aggregation. ONE WAVE PER EDGE; each lane owns 4 features:
//   agg[col][f] += dis[row]*w*dis[col] * t[row][f]
//   msg[col][f] += w * h[row][f]
// Gathers become global_load_b128; edge scalars are wave-broadcast loads.
// ---------------------------------------------------------------------------
__global__ void __launch_bounds__(256)
edge_agg_kernel(const int* __restrict__ ei, const float* __restrict__ ew,
                const float* __restrict__ dis, const float* __restrict__ h,
                const float* __restrict__ t, float* __restrict__ agg,
                float* __restrict__ msg, int E) {
    const int e = blockIdx.x * 8 + (threadIdx.x >> 5);
    if (e >= E) return;
    const int lane = threadIdx.x & 31;
    const int f0 = lane * 4;

    const int r = ei[e];
    const int c = ei[E + e];
    const float w = ew[e];
    const float nrm = dis[r] * w * dis[c];

    const float4 tv = *(const float4*)(t + (size_t)r * HID + f0);
    const float4 hv = *(const float4*)(h + (size_t)r * HID + f0);

    float* aggp = agg + (size_t)c * HID + f0;
    float* msgp = msg + (size_t)c * HID + f0;
    atomicAdd(aggp + 0, nrm * tv.x);
    atomicAdd(aggp + 1, nrm * tv.y);
    atomicAdd(aggp + 2, nrm * tv.z);
    atomicAdd(aggp + 3, nrm * tv.w);
    atomicAdd(msgp + 0, w * hv.x);
    atomicAdd(msgp + 1, w * hv.y);
    atomicAdd(msgp + 2, w * hv.z);
    atomicAdd(msgp + 3, w * hv.w);
}

// ---------------------------------------------------------------------------
// Kernel 5: finalize. Stage h-tile and mean-tile in LDS (float4); 3 WMMA
// GEMMs per 16-row tile: hr = h@W_root, hs = h@W_sage_r, sl = mean@W_sage_l;
// then activations and concat store into out[N, 256].
// ---------------------------------------------------------------------------
__global__ void __launch_bounds__(256)
finalize_kernel(const float* __restrict__ h, const float* __restrict__ agg,
                const float* __restrict__ msg, const float* __restrict__ cnt,
                const float* __restrict__ Wroot, const float* __restrict__ armab,
                const float* __restrict__ Wl, const float* __restrict__ bl,
                const float* __restrict__ Wr, float* __restrict__ out, int N) {
    __shared__ float hTile[16 * HID];
    __shared__ float meanTile[16 * HID];
    const int tid = threadIdx.x;
    const int m0 = blockIdx.x * 16;

    for (int i = tid; i < 16 * HID / 4; i += 256) {
        const int rr = i >> 5;
        const int ff4 = i & 31;
        int grow = m0 + rr;
        if (grow >= N) grow = N - 1;
        ((float4*)hTile)[i] =
            ((const float4*)(h + (size_t)grow * HID))[ff4];
        const float inv = 1.0f / fmaxf(cnt[grow], 1.0f);
        float4 m = ((const float4*)(msg + (size_t)grow * HID))[ff4];
        m.x *= inv; m.y *= inv; m.z *= inv; m.w *= inv;
        ((float4*)meanTile)[i] = m;
    }
    __syncthreads();

    const int wave = tid >> 5;
    const int lane = tid & 31;
    const int l = lane & 15;
    const int half = lane >> 4;
    const int n0 = wave * 16;

    v8f hr = wave_gemm128(hTile, l, Wroot + n0, l, half);
    v8f hs = wave_gemm128(hTile, l, Wr + n0, l, half);
    v8f sl = wave_gemm128(meanTile, l, Wl + n0, l, half);

    const float ab = armab[n0 + l];
    const float sb = bl[n0 + l];
#pragma unroll
    for (int r = 0; r < 8; ++r) {
        const int grow = m0 + 8 * half + r;
        if (grow >= N) continue;
        // ARMA branch: relu(agg + h@root + b), then leaky, then elu
        float a = agg[(size_t)grow * HID + n0 + l] + hr[r] + ab;
        a = fmaxf(a, 0.0f);
        float h1 = (a > 0.0f) ? a : 0.01f * a;
        float o1 = (h1 > 0.0f) ? h1 : (expf(h1) - 1.0f);
        // SAGE branch: mean@l_w + b + h@r_w, leaky, elu
        float s = sl[r] + sb + hs[r];
        float h2 = (s > 0.0f) ? s : 0.01f * s;
        float o2 = (h2 > 0.0f) ? h2 : (expf(h2) - 1.0f);
        out[(size_t)grow * (2 * OUTD) + n0 + l] = o1;
        out[(size_t)grow * (2 * OUTD) + OUTD + n0 + l] = o2;
    }
}

// ---------------------------------------------------------------------------
extern "C" void kernel_launch(void* const* d_in, const int* in_sizes, int n_in,
                              void* d_out, int out_size, void* d_ws, size_t ws_size,
                              hipStream_t stream) {
    const float* x     = (const float*)d_in[0];
    const int*   ei    = (const int*)d_in[1];     // [2, E]
    const float* ew    = (const float*)d_in[2];   // [E]
    const float* Wpre  = (const float*)d_in[3];
    const float* bpre  = (const float*)d_in[4];
    const float* Winit = (const float*)d_in[5];
    const float* Wroot = (const float*)d_in[6];
    const float* armab = (const float*)d_in[7];
    const float* Wl    = (const float*)d_in[8];
    const float* bl    = (const float*)d_in[9];
    const float* Wr    = (const float*)d_in[10];
    float* out = (float*)d_out;

    const int N = in_sizes[0] / HID;
    const int E = in_sizes[2];

    float* ws  = (float*)d_ws;
    float* h   = ws; ws += (size_t)N * HID;
    float* t   = ws; ws += (size_t)N * HID;
    float* agg = ws; ws += (size_t)N * HID;   // contiguous with msg
    float* msg = ws; ws += (size_t)N * HID;
    float* deg = ws; ws += N;                 // contiguous with cnt
    float* cnt = ws; ws += N;
    float* dis = ws; ws += N;

    // Zero the accumulators each call (deterministic; graph-capturable memsets).
    hipMemsetAsync(agg, 0, (size_t)2 * N * HID * sizeof(float), stream);
    hipMemsetAsync(deg, 0, (size_t)2 * N * sizeof(float), stream);

    const int nTiles = (N + 15) / 16;

    gemm_pre_init<<<nTiles, 256, 0, stream>>>(x, Wpre, bpre, Winit, h, t, N);

    deg_cnt_kernel<<<(E + 255) / 256, 256, 0, stream>>>(ei, ew, deg, cnt, E);
    dis_kernel<<<(N + 255) / 256, 256, 0, stream>>>(deg, dis, N);

    edge_agg_kernel<<<(E + 7) / 8, 256, 0, stream>>>(ei, ew, dis, h, t, agg,
                                                     msg, E);

    finalize_kernel<<<nTiles, 256, 0, stream>>>(h, agg, msg, cnt, Wroot, armab,
                                                Wl, bl, Wr, out, N);
}